// GATNet_26044681683301
// MI455X (gfx1250) — compile-verified
//
#include <hip/hip_runtime.h>
#include <hip/hip_bf16.h>

typedef __attribute__((ext_vector_type(2))) float v2f;
typedef __attribute__((ext_vector_type(8))) float v8f;

#define NEG_SLOPE 0.01f
#define BN_EPS 1e-5f
#define DMODEL 128
#define NGRAPH 256

// ---------- order-preserving float<->uint for atomic max ----------
__device__ __forceinline__ unsigned f2ord(float f) {
  unsigned u = __float_as_uint(f);
  return (u >> 31) ? ~u : (u | 0x80000000u);
}
__device__ __forceinline__ float ord2f(unsigned u) {
  return (u >> 31) ? __uint_as_float(u & 0x7FFFFFFFu) : __uint_as_float(~u);
}

__global__ void k_zero(float* p, long n) {
  long i = (long)blockIdx.x * blockDim.x + threadIdx.x;
  long stride = (long)gridDim.x * blockDim.x;
  for (; i < n; i += stride) p[i] = 0.f;
}

// ---------------------------------------------------------------------------
// C[N,128] = A[N,128] @ B + bias   (fp32 WMMA 16x16x4)
// B is logical [128,128]; physical element (d,c) lives at  d*sd + (c/Oc)*sh + c%Oc
// (absorbs the fcW [H,Din,O] head permutation into LDS staging).
// LDS layout is K-pair interleaved:  (k,c) -> (k>>1)*256 + c*2 + (k&1)
// so each B fragment is ONE aligned ds_load_b64 directly into the WMMA pair.
// Block = 128 threads = 4 waves; each wave owns a 16-row x 128-col tile.
// ---------------------------------------------------------------------------
__global__ __launch_bounds__(128) void k_gemm_wmma(
    const float* __restrict__ A, const float* __restrict__ B,
    const float* __restrict__ bias, float* __restrict__ C,
    int N, int sd, int sh, int Oc)
{
  __shared__ float Bs[64 * 128];           // 32 KB K-panel of B (interleaved)
  const int lane = threadIdx.x & 31;
  const int wave = threadIdx.x >> 5;
  const int row0 = (blockIdx.x * 4 + wave) * 16;
  const int m = lane & 15;                 // row within A frag / col within B frag
  const int khalf = (lane >> 4) * 2;       // lanes 16-31 hold K+2,K+3

  v8f acc[8] = {};

  for (int kb = 0; kb < 128; kb += 64) {
    // prefetch this wave's A rows for the panel we're about to consume
    if (row0 < N)
      __builtin_prefetch(A + (long)(row0 + m) * DMODEL + kb + khalf, 0, 0);
    __syncthreads();
    for (int idx = threadIdx.x; idx < 64 * 128; idx += blockDim.x) {
      int kl = idx >> 7;                   // 0..63 (local K)
      int c  = idx & 127;
      int d  = kb + kl;
      Bs[(kl >> 1) * 256 + c * 2 + (kl & 1)] = B[d * sd + (c / Oc) * sh + (c % Oc)];
    }
    __syncthreads();
    if (row0 < N) {                        // wave-uniform: EXEC stays all-ones
      #pragma unroll 4
      for (int k0 = 0; k0 < 64; k0 += 4) {
        const float* ap = A + (long)(row0 + m) * DMODEL + kb + k0 + khalf;
        v2f a; a.x = ap[0]; a.y = ap[1];
        const int krow = (k0 + khalf) >> 1;          // even k -> one b64 word
        #pragma unroll
        for (int t = 0; t < 8; ++t) {
          v2f b = *(const v2f*)(Bs + krow * 256 + (t * 16 + m) * 2);
          acc[t] = __builtin_amdgcn_wmma_f32_16x16x4_f32(
              false, a, false, b, (short)0, acc[t], false, false);
        }
      }
    }
  }
  if (row0 < N) {
    #pragma unroll
    for (int t = 0; t < 8; ++t) {
      #pragma unroll
      for (int i = 0; i < 8; ++i) {
        int r = row0 + i + (lane >> 4) * 8;  // C/D layout: vgpr i -> row i / i+8
        int c = t * 16 + m;
        float v = acc[t][i];
        if (bias) v += bias[c];
        C[(long)r * DMODEL + c] = v;
      }
    }
  }
}

// el[n,h] = z[n,h,:].aS[h,:] ; er likewise
__global__ void k_attn_coef(const float* __restrict__ z, const float* __restrict__ aS,
                            const float* __restrict__ aD, float* __restrict__ el,
                            float* __restrict__ er, int N, int H, int O)
{
  int i = blockIdx.x * blockDim.x + threadIdx.x;
  int total = N * H;
  int stride = gridDim.x * blockDim.x;
  for (; i < total; i += stride) {
    int n = i / H, h = i - n * H;
    const float* zp = z + (long)n * DMODEL + h * O;
    float sl = 0.f, sr = 0.f;
    for (int o = 0; o < O; ++o) { float zz = zp[o]; sl += zz * aS[h*O+o]; sr += zz * aD[h*O+o]; }
    el[i] = sl; er[i] = sr;
  }
}

__device__ __forceinline__ float edge_e(const float* el, const float* er,
                                        int s, int d, int H, int h) {
  float v = el[s * H + h] + er[d * H + h];
  return (v > 0.f) ? v : NEG_SLOPE * v;
}

__global__ void k_edge_max(const int* __restrict__ src, const int* __restrict__ dst,
                           const float* __restrict__ el, const float* __restrict__ er,
                           unsigned* __restrict__ emax, int E, int H)
{
  int e = blockIdx.x * blockDim.x + threadIdx.x;
  int stride = gridDim.x * blockDim.x;
  for (; e < E; e += stride) {
    int s = src[e], d = dst[e];
    for (int h = 0; h < H; ++h)
      atomicMax(&emax[d * H + h], f2ord(edge_e(el, er, s, d, H, h)));
  }
}

__global__ void k_edge_denom(const int* __restrict__ src, const int* __restrict__ dst,
                             const float* __restrict__ el, const float* __restrict__ er,
                             const unsigned* __restrict__ emax, float* __restrict__ denom,
                             int E, int H)
{
  int e = blockIdx.x * blockDim.x + threadIdx.x;
  int stride = gridDim.x * blockDim.x;
  for (; e < E; e += stride) {
    int s = src[e], d = dst[e];
    for (int h = 0; h < H; ++h) {
      float v = edge_e(el, er, s, d, H, h);
      atomicAdd(&denom[d * H + h], __expf(v - ord2f(emax[d * H + h])));
    }
  }
}

// agg[dst, :] += alpha * z[src, :]  (4 threads per edge, 32 cols each; alpha recomputed)
__global__ void k_edge_scatter(const int* __restrict__ src, const int* __restrict__ dst,
                               const float* __restrict__ el, const float* __restrict__ er,
                               const unsigned* __restrict__ emax, const float* __restrict__ denom,
                               const float* __restrict__ z, float* __restrict__ agg,
                               int E, int H, int O)
{
  long i = (long)blockIdx.x * blockDim.x + threadIdx.x;
  long total = (long)E * 4;
  long stride = (long)gridDim.x * blockDim.x;
  for (; i < total; i += stride) {
    int e = (int)(i >> 2);
    int cbase = (int)(i & 3) * 32;
    int s = src[e], d = dst[e];
    int hcur = -1; float alpha = 0.f;
    #pragma unroll
    for (int q = 0; q < 8; ++q) {
      int c = cbase + q * 4;
      int h = c / O;
      if (h != hcur) {
        hcur = h;
        float v = edge_e(el, er, s, d, H, h);
        alpha = __expf(v - ord2f(emax[d * H + h])) / denom[d * H + h];
      }
      const float4 zz = *(const float4*)(z + (long)s * DMODEL + c);
      float* ap = agg + (long)d * DMODEL + c;
      atomicAdd(ap + 0, alpha * zz.x);
      atomicAdd(ap + 1, alpha * zz.y);
      atomicAdd(ap + 2, alpha * zz.z);
      atomicAdd(ap + 3, alpha * zz.w);
    }
  }
}

// in-place agg *= snorm; per-channel sum/sumsq partials -> gstats[0:128]/[128:256]
__global__ __launch_bounds__(128) void k_scale_stats(float* __restrict__ agg,
                                                     const float* __restrict__ snorm,
                                                     float* __restrict__ gstats, int N)
{
  const int t = threadIdx.x;               // channel 0..127
  const int n0 = blockIdx.x * 256;
  float s = 0.f, ss = 0.f;
  for (int r = 0; r < 256; ++r) {
    int n = n0 + r;
    if (n < N) {
      float x = agg[(long)n * DMODEL + t] * snorm[n];
      agg[(long)n * DMODEL + t] = x;
      s += x; ss += x * x;
    }
  }
  atomicAdd(&gstats[t], s);
  atomicAdd(&gstats[128 + t], ss);
}

// h_next = h_in + elu( (x-mu)*rsqrt(var+eps)*gamma + beta ), written over agg
__global__ void k_bn_elu_res(float* __restrict__ agg, const float* __restrict__ hin,
                             const float* __restrict__ gstats,
                             const float* __restrict__ gamma, const float* __restrict__ beta,
                             int N)
{
  long i = (long)blockIdx.x * blockDim.x + threadIdx.x;
  long total = (long)N * DMODEL;
  long stride = (long)gridDim.x * blockDim.x;
  float invN = 1.f / (float)N;
  for (; i < total; i += stride) {
    int c = (int)(i & 127);
    float mu = gstats[c] * invN;
    float var = gstats[128 + c] * invN - mu * mu;
    float y = (agg[i] - mu) * rsqrtf(var + BN_EPS) * gamma[c] + beta[c];
    y = (y > 0.f) ? y : (__expf(y) - 1.f);
    agg[i] = hin[i] + y;
  }
}

__global__ void k_graph_accum(const float* __restrict__ h, const int* __restrict__ gid,
                              float* __restrict__ hg, float* __restrict__ counts, int N)
{
  long i = (long)blockIdx.x * blockDim.x + threadIdx.x;
  long total = (long)N * DMODEL;
  long stride = (long)gridDim.x * blockDim.x;
  for (; i < total; i += stride) {
    int n = (int)(i >> 7), c = (int)(i & 127);
    int g = gid[n];
    atomicAdd(&hg[(long)g * DMODEL + c], h[i]);
    if (c == 0) atomicAdd(&counts[g], 1.f);
  }
}

__global__ __launch_bounds__(128) void k_mlp(
    const float* __restrict__ hg, const float* __restrict__ counts,
    const float* __restrict__ w1, const float* __restrict__ b1,
    const float* __restrict__ w2, const float* __restrict__ b2,
    const float* __restrict__ w3, const float* __restrict__ b3,
    float* __restrict__ out)
{
  __shared__ float row[128];
  __shared__ float y1[64];
  __shared__ float y2[32];
  int g = blockIdx.x, t = threadIdx.x;
  float cnt = counts[g]; if (cnt < 1.f) cnt = 1.f;
  row[t] = hg[(long)g * DMODEL + t] / cnt;
  __syncthreads();
  if (t < 64) { float a = b1[t]; for (int k = 0; k < 128; ++k) a += row[k] * w1[k*64+t]; y1[t] = fmaxf(a, 0.f); }
  __syncthreads();
  if (t < 32) { float a = b2[t]; for (int k = 0; k < 64; ++k) a += y1[k] * w2[k*32+t]; y2[t] = fmaxf(a, 0.f); }
  __syncthreads();
  if (t < 10) { float a = b3[t]; for (int k = 0; k < 32; ++k) a += y2[k] * w3[k*10+t]; out[g*10+t] = a; }
}

extern "C" void kernel_launch(void* const* d_in, const int* in_sizes, int n_in,
                              void* d_out, int out_size, void* d_ws, size_t ws_size,
                              hipStream_t stream)
{
  (void)n_in; (void)out_size; (void)ws_size;
  const float* nodes = (const float*)d_in[0];
  const float* snorm = (const float*)d_in[1];
  const float* embW  = (const float*)d_in[2];
  const float* embB  = (const float*)d_in[3];
  const float* fcW   = (const float*)d_in[4];
  const float* aS    = (const float*)d_in[5];
  const float* aD    = (const float*)d_in[6];
  const float* gam   = (const float*)d_in[7];
  const float* bet   = (const float*)d_in[8];
  const float* fcWL  = (const float*)d_in[9];
  const float* aSL   = (const float*)d_in[10];
  const float* aDL   = (const float*)d_in[11];
  const float* gL    = (const float*)d_in[12];
  const float* bL    = (const float*)d_in[13];
  const float* w1    = (const float*)d_in[14];
  const float* b1    = (const float*)d_in[15];
  const float* w2    = (const float*)d_in[16];
  const float* b2    = (const float*)d_in[17];
  const float* w3    = (const float*)d_in[18];
  const float* b3    = (const float*)d_in[19];
  const int* src     = (const int*)d_in[20];
  const int* dst     = (const int*)d_in[21];
  const int* gid     = (const int*)d_in[22];

  const int N = in_sizes[1];
  const int E = in_sizes[20];
  const int Lmid = in_sizes[4] / (8 * 128 * 16);   // = 3

  // ---- carve workspace (all regions 256B aligned) ----
  char* wp = (char*)d_ws;
  auto carve = [&](size_t bytes) -> float* {
    float* p = (float*)wp;
    wp += (bytes + 255) & ~(size_t)255;
    return p;
  };
  float* bufA   = carve((size_t)N * 128 * 4);
  float* bufB   = carve((size_t)N * 128 * 4);
  float* bufC   = carve((size_t)N * 128 * 4);
  float* el     = carve((size_t)N * 8 * 4);
  float* er     = carve((size_t)N * 8 * 4);
  unsigned* emx = (unsigned*)carve((size_t)N * 8 * 4);
  float* denom  = carve((size_t)N * 8 * 4);
  float* gstats = carve(256 * 4);
  float* hg     = carve((size_t)NGRAPH * 128 * 4);
  float* counts = carve(NGRAPH * 4);

  float* h = bufA; float* z = bufB; float* agg = bufC;

  const int gemmBlocks = (((N + 15) / 16) + 3) / 4;

  // embedding: h = X @ embW + embB
  k_gemm_wmma<<<gemmBlocks, 128, 0, stream>>>(nodes, embW, embB, h, N, 128, 0, 128);

  auto layer = [&](const float* W, int sd, int sh, int Oc,
                   const float* aS_, const float* aD_,
                   const float* g_, const float* b_, int Hh, int Oo)
  {
    k_zero<<<2048, 256, 0, stream>>>(agg, (long)N * 128);
    k_zero<<<256, 256, 0, stream>>>((float*)emx, (long)N * Hh);
    k_zero<<<256, 256, 0, stream>>>(denom, (long)N * Hh);
    k_zero<<<1, 256, 0, stream>>>(gstats, 256);

    k_gemm_wmma<<<gemmBlocks, 128, 0, stream>>>(h, W, nullptr, z, N, sd, sh, Oc);
    int tot = N * Hh;
    k_attn_coef<<<(tot + 255) / 256, 256, 0, stream>>>(z, aS_, aD_, el, er, N, Hh, Oo);
    k_edge_max<<<(E + 255) / 256, 256, 0, stream>>>(src, dst, el, er, emx, E, Hh);
    k_edge_denom<<<(E + 255) / 256, 256, 0, stream>>>(src, dst, el, er, emx, denom, E, Hh);
    long sc = (long)E * 4;
    k_edge_scatter<<<(int)((sc + 255) / 256), 256, 0, stream>>>(src, dst, el, er, emx, denom,
                                                                z, agg, E, Hh, Oo);
    k_scale_stats<<<(N + 255) / 256, 128, 0, stream>>>(agg, snorm, gstats, N);
    k_bn_elu_res<<<2048, 256, 0, stream>>>(agg, h, gstats, g_, b_, N);

    float* oldh = h; h = agg; agg = z; z = oldh;   // rotate buffers
  };

  for (int l = 0; l < Lmid; ++l)
    layer(fcW + (size_t)l * 8 * 128 * 16, /*sd=*/16, /*sh=*/2048, /*Oc=*/16,
          aS + l * 128, aD + l * 128, gam + l * 128, bet + l * 128, /*H=*/8, /*O=*/16);
  layer(fcWL, /*sd=*/128, /*sh=*/0, /*Oc=*/128, aSL, aDL, gL, bL, /*H=*/1, /*O=*/128);

  k_zero<<<64, 256, 0, stream>>>(hg, (long)NGRAPH * 128);
  k_zero<<<1, 256, 0, stream>>>(counts, NGRAPH);
  k_graph_accum<<<4096, 256, 0, stream>>>(h, gid, hg, counts, N);
  k_mlp<<<NGRAPH, 128, 0, stream>>>(hg, counts, w1, b1, w2, b2, w3, b3, (float*)d_out);
}